// AdditiveAttention_68788196213342
// MI455X (gfx1250) — compile-verified
//
#include <hip/hip_runtime.h>

typedef float v2f __attribute__((ext_vector_type(2)));
typedef float v8f __attribute__((ext_vector_type(8)));

#define NEGV (-1000000.0f)

#if __has_builtin(__builtin_amdgcn_tanhf)
#define TANHF __builtin_amdgcn_tanhf
#elif __has_builtin(__builtin_amdgcn_tanh_f32)
#define TANHF __builtin_amdgcn_tanh_f32
#else
#define TANHF tanhf
#endif

// ---------------------------------------------------------------------------
// Batched FP32 GEMM, one wave (32 lanes) per 16x16 output tile, accumulated
// in fp32 via V_WMMA_F32_16X16X4_F32.  (Round-1 verified: wmma lowers, loads
// merge to global_load_b64.)
// ---------------------------------------------------------------------------
__global__ __launch_bounds__(32) void gemm16_wmma_f32(
    const float* __restrict__ A, const float* __restrict__ Bm,
    float* __restrict__ C, int M, int N, int K,
    long sA, long sB, long sC)
{
    const int lane = threadIdx.x;
    const int tilesN = N >> 4;
    const int m0 = (blockIdx.x / tilesN) << 4;
    const int n0 = (blockIdx.x % tilesN) << 4;

    const float* Ab = A  + (long)blockIdx.y * sA;
    const float* Bb = Bm + (long)blockIdx.y * sB;
    float*       Cb = C  + (long)blockIdx.y * sC;

    const int half = lane >> 4;
    const int l15  = lane & 15;

    const float* arow = Ab + (long)(m0 + l15) * K;

    v8f acc = {};
    for (int kk = 0; kk < K; kk += 4) {
        const int ks = kk + 2 * half;
        v2f a, b;
        a.x = arow[ks];
        a.y = arow[ks + 1];
        b.x = Bb[(long)(ks)     * N + n0 + l15];
        b.y = Bb[(long)(ks + 1) * N + n0 + l15];
        acc = __builtin_amdgcn_wmma_f32_16x16x4_f32(
            false, a, false, b, (short)0, acc, false, false);
    }

#pragma unroll
    for (int g = 0; g < 8; ++g) {
        Cb[(long)(m0 + g + 8 * half) * N + n0 + l15] = acc[g];
    }
}

// ---------------------------------------------------------------------------
// Async global->LDS 16-byte copy (CDNA5 GLOBAL_LOAD_ASYNC_TO_LDS_B128, GV
// mode: vdst = LDS byte offset, vaddr = 64-bit global address).  Tracked by
// ASYNCcnt; completion enforced with s_wait_asynccnt.
// ---------------------------------------------------------------------------
__device__ __forceinline__ void async_copy16(unsigned lds_off, const float* g)
{
    asm volatile("global_load_async_to_lds_b128 %0, %1, off"
                 :: "v"(lds_off), "v"(g) : "memory");
}

__device__ __forceinline__ void wait_async0()
{
    asm volatile("s_wait_asynccnt 0x0" ::: "memory");
}

// ---------------------------------------------------------------------------
// Fused additive score + masked softmax.
// One 256-thread block (8 waves) per (b, 8 q-rows): 128 blocks total, so
// kproj is streamed only 8x per batch element (64 MB total ~= 2.7 us @HBM)
// instead of 64x.
//
// Per 32-row kproj tile (staged in LDS, double-buffered via async DMA):
//   lane owns one (q,k) pair per wave: q = lane>>2 (8 q), k = wave*4+(lane&3)
//   -> 8 waves x 32 lanes cover all 8 q x 32 k dots, no shuffle reductions.
// LDS strides padded (260 / 520 floats) for bank-conflict-free reads while
// keeping rows 16B-aligned for the B128 async writes.
// Softmax: wave w owns q-row w -> wave-local __shfl_xor reductions only.
// LDS: 2*32*260*4 + 8*260*4 + 256*4 + 8*520*4 ~= 93 KB  (WGP has 320 KB)
// ---------------------------------------------------------------------------
#define QPB     8
#define TILE_K  32
#define NTILES  16          // 512 / 32
#define KT_STR  260         // padded row stride (floats), 16B aligned
#define SC_STR  520

__global__ __launch_bounds__(256) void additive_score_softmax(
    const float* __restrict__ qproj,      // (B*Q, 256)
    const float* __restrict__ kproj,      // (B*512, 256)
    const float* __restrict__ wv,         // (256,)
    const int*   __restrict__ valid_lens, // (B,)
    float*       __restrict__ attn)       // (B*Q, 512)
{
    const int b    = blockIdx.x >> 3;         // 16 batches
    const int q0   = (blockIdx.x & 7) * QPB;  // 8 q-groups of 8
    const int t    = threadIdx.x;
    const int wave = t >> 5;
    const int lane = t & 31;

    __shared__ __align__(16) float ktile[2][TILE_K][KT_STR];
    __shared__ __align__(16) float qv[QPB][KT_STR];
    __shared__ __align__(16) float wvs[256];
    __shared__ float sc[QPB][SC_STR];

    const unsigned kt_base = (unsigned)(unsigned long long)(&ktile[0][0][0]);
    const unsigned kt_bufsz = (unsigned)(TILE_K * KT_STR * 4);
    const float* kbase = kproj + (long)b * 512 * 256;

    // --- start DMA of tile 0 while we load qv/wvs the normal way ----------
    {
#pragma unroll
        for (int j = 0; j < 8; ++j) {
            const int v    = t + 256 * j;   // float4 index within 32x256 tile
            const int grow = v >> 6;
            const int gc4  = v & 63;
            async_copy16(kt_base + (unsigned)(grow * KT_STR + gc4 * 4) * 4u,
                         kbase + grow * 256 + gc4 * 4);
        }
    }
#pragma unroll
    for (int j = 0; j < QPB; ++j)
        qv[j][t] = qproj[(long)(b * 64 + q0 + j) * 256 + t];
    wvs[t] = wv[t];

    const int q  = lane >> 2;               // 0..7
    const int ki = lane & 3;                // 0..3
    const int kk = wave * 4 + ki;           // tile row 0..31 owned by lane
    const float* qr = &qv[q][0];

    for (int tI = 0; tI < NTILES; ++tI) {
        const int buf = tI & 1;
        wait_async0();
        __syncthreads();                    // tile tI resident; prev reads done

        if (tI + 1 < NTILES) {              // prefetch next tile into other buf
            const float* src = kbase + (long)(tI + 1) * TILE_K * 256;
            const unsigned lb = kt_base + (unsigned)(buf ^ 1) * kt_bufsz;
#pragma unroll
            for (int j = 0; j < 8; ++j) {
                const int v    = t + 256 * j;
                const int grow = v >> 6;
                const int gc4  = v & 63;
                async_copy16(lb + (unsigned)(grow * KT_STR + gc4 * 4) * 4u,
                             src + grow * 256 + gc4 * 4);
            }
        }

        // dot product for this lane's (q, k) pair
        const float* kr = &ktile[buf][kk][0];
        float s = 0.0f;
#pragma unroll 8
        for (int h = 0; h < 256; h += 4) {
            const float4 k4 = *(const float4*)(kr + h);
            const float4 q4 = *(const float4*)(qr + h);
            const float4 w4 = *(const float4*)(&wvs[h]);
            s += w4.x * TANHF(q4.x + k4.x);
            s += w4.y * TANHF(q4.y + k4.y);
            s += w4.z * TANHF(q4.z + k4.z);
            s += w4.w * TANHF(q4.w + k4.w);
        }
        sc[q][tI * TILE_K + kk] = s;
    }
    __syncthreads();

    // --- masked softmax: wave w owns q-row w ------------------------------
    const int vl = valid_lens[b];
    float x[16];
    float mx = NEGV;
#pragma unroll
    for (int i = 0; i < 16; ++i) {
        const int k = lane + 32 * i;
        x[i] = (k < vl) ? sc[wave][k] : NEGV;
        mx = fmaxf(mx, x[i]);
    }
#pragma unroll
    for (int off = 16; off > 0; off >>= 1)
        mx = fmaxf(mx, __shfl_xor(mx, off, 32));

    float ssum = 0.0f;
#pragma unroll
    for (int i = 0; i < 16; ++i) {
        x[i] = __expf(x[i] - mx);
        ssum += x[i];
    }
#pragma unroll
    for (int off = 16; off > 0; off >>= 1)
        ssum += __shfl_xor(ssum, off, 32);
    const float inv = 1.0f / ssum;

    float* arow = attn + (long)(b * 64 + q0 + wave) * 512;
#pragma unroll
    for (int i = 0; i < 16; ++i)
        arow[lane + 32 * i] = x[i] * inv;
}

// ---------------------------------------------------------------------------
// Launcher.  B=16, Q=64, KV=512, QS=KS=H=DV=256.
// Workspace (floats): qproj[1024*256] | kproj[8192*256] | attn[1024*512]
// ---------------------------------------------------------------------------
extern "C" void kernel_launch(void* const* d_in, const int* in_sizes, int n_in,
                              void* d_out, int out_size, void* d_ws, size_t ws_size,
                              hipStream_t stream)
{
    const float* queries    = (const float*)d_in[0];
    const float* keys       = (const float*)d_in[1];
    const float* values     = (const float*)d_in[2];
    const int*   valid_lens = (const int*)  d_in[3];
    const float* Wq         = (const float*)d_in[4];
    const float* Wk         = (const float*)d_in[5];
    const float* wv         = (const float*)d_in[6];
    float*       out        = (float*)d_out;

    float* ws    = (float*)d_ws;
    float* qproj = ws;
    float* kproj = ws + 1024 * 256;
    float* attn  = ws + 1024 * 256 + 8192 * 256;

    const dim3 wave(32);

    gemm16_wmma_f32<<<dim3((1024 / 16) * (256 / 16), 1), wave, 0, stream>>>(
        queries, Wq, qproj, 1024, 256, 256, 0L, 0L, 0L);

    gemm16_wmma_f32<<<dim3((8192 / 16) * (256 / 16), 1), wave, 0, stream>>>(
        keys, Wk, kproj, 8192, 256, 256, 0L, 0L, 0L);

    additive_score_softmax<<<dim3(128), dim3(256), 0, stream>>>(
        qproj, kproj, wv, valid_lens, attn);

    gemm16_wmma_f32<<<dim3((64 / 16) * (256 / 16), 16), wave, 0, stream>>>(
        attn, values, out, 64, 256, 512,
        64L * 512, 512L * 256, 64L * 256);
}